// TrainableGeneSetLayer_43121471652195
// MI455X (gfx1250) — compile-verified
//
#include <hip/hip_runtime.h>

#define G_GENES 20000
#define K_SETS  128
#define B_BATCH 16
#define GTILE   200   // 20000 / 200 = 100 exact tiles -> no tail guards in hot loop

typedef __attribute__((ext_vector_type(2))) float v2f;
typedef __attribute__((ext_vector_type(8))) float v8f;

__device__ __forceinline__ float sigmoidf(float x) {
    return 1.0f / (1.0f + __expf(-x));
}

// Fast x^0.25 : input already clamped to [1e-8, 1e4] -> raw v_sqrt_f32 is safe.
__device__ __forceinline__ float fourth_root(float x) {
    return __builtin_amdgcn_sqrtf(__builtin_amdgcn_sqrtf(x));
}

// ---------------------------------------------------------------------------
// Pass 1: per-set mean of sigmoid(membership)  (128 reductions over 20000)
// ---------------------------------------------------------------------------
__global__ __launch_bounds__(256)
void k_avg(const float* __restrict__ M, float* __restrict__ avg) {
    const int k = blockIdx.x;
    const float* row = M + (size_t)k * G_GENES;
    float s = 0.f;
    for (int j = threadIdx.x; j < G_GENES; j += 256) s += sigmoidf(row[j]);
    __shared__ float red[256];
    red[threadIdx.x] = s;
    __syncthreads();
    for (int off = 128; off > 0; off >>= 1) {
        if (threadIdx.x < off) red[threadIdx.x] += red[threadIdx.x + off];
        __syncthreads();
    }
    if (threadIdx.x == 0) avg[k] = red[0] * (1.0f / (float)G_GENES);
}

// ---------------------------------------------------------------------------
// Pass 2: build down-weighted indicators, TRANSPOSED to indT[G][K] so that a
// gene's 128 set-values are one contiguous 512B line (coalesced per wave).
// ---------------------------------------------------------------------------
__global__ __launch_bounds__(256)
void k_build(const float* __restrict__ M, const float* __restrict__ avg,
             float* __restrict__ indT) {
    __shared__ float tile[32][33];
    const int j0 = blockIdx.x * 32;   // gene tile (625 exact)
    const int k0 = blockIdx.y * 32;   // set tile  (4 exact)
    for (int i = threadIdx.y; i < 32; i += 8) {
        const int k = k0 + i, j = j0 + threadIdx.x;
        const float s = sigmoidf(M[(size_t)k * G_GENES + j]);
        const float a = avg[k];
        tile[i][threadIdx.x] = (s < a * 0.3f) ? s * 0.01f : s;
    }
    __syncthreads();
    for (int i = threadIdx.y; i < 32; i += 8) {
        const int j = j0 + i, k = k0 + threadIdx.x;
        indT[(size_t)j * K_SETS + k] = tile[threadIdx.x][i];
    }
}

// ---------------------------------------------------------------------------
// Zero the 4 x (B*K) float accumulators (atomics accumulate across the run).
// ---------------------------------------------------------------------------
__global__ void k_zero(float* __restrict__ acc, int n) {
    const int i = blockIdx.x * blockDim.x + threadIdx.x;
    if (i < n) acc[i] = 0.f;
}

// ---------------------------------------------------------------------------
// Main pass. Block = (gene-tile, batch b), 8 waves; wave w owns sets
// [16w, 16w+16). Nonlinearity computed per-lane directly in the
// V_WMMA_F32_16X16X4_F32 A-fragment layout; WMMA performs the exact f32
// contraction against B = [ones | (G-g)] columns.
//   A layout (16x4 f32): lanes 0-15 = M rows; VGPR0 holds K = 2*half,
//                        VGPR1 holds K = 2*half+1  (half = lane>=16).
//   B layout (4x16 f32): lane gives column N; VGPR0 = row 2*half,
//                        VGPR1 = row 2*half+1.
//   D layout (16x16):    lane gives column N; VGPR v = row v + 8*half.
// ---------------------------------------------------------------------------
__global__ __launch_bounds__(256)
void k_main(const float* __restrict__ R, const int* __restrict__ S,
            const float* __restrict__ indT, float* __restrict__ acc) {
    // Interleaved staging: sbuf[2g] = S[b,g]*K (pre-scaled row offset),
    //                      sbuf[2g+1] = bits of R[b, S[b,g]].
    __shared__ int sbuf[GTILE * 2];

    const int b  = blockIdx.y;
    const int g0 = blockIdx.x * GTILE;

    {
        const int t = threadIdx.x;
        if (t < GTILE) {
            const int j = S[b * G_GENES + g0 + t];
            const float r = R[b * G_GENES + j];
            sbuf[2 * t]     = j * K_SETS;
            sbuf[2 * t + 1] = __float_as_int(r);
        }
    }
    __syncthreads();

    const int lane = threadIdx.x & 31;
    const int m    = lane & 15;            // A row (set-in-tile) AND D/B column
    const int half = lane >> 4;            // selects K pair {0,1} vs {2,3}
    const int k0   = (threadIdx.x >> 5) * 16;
    const int kcol = k0 + m;

    // B-fragment column selectors: col0 -> plain sum, col1 -> sum of (G-g).
    const float sel0 = (m == 0) ? 1.f : 0.f;
    const float sel1 = (m == 1) ? 1.f : 0.f;
    // coefficient (G - g) for this lane's two B rows, stepped per 4-gene chunk
    float coef0 = (float)(G_GENES - (g0 + half * 2));
    float coef1 = coef0 - 1.f;

    v8f dpos = {};   // cols: 0 -> P1, 1 -> P2
    v8f dneg = {};   // cols: 0 -> N1, 1 -> N2

    #pragma unroll 2
    for (int c = 0; c < GTILE / 4; ++c) {
        const int idx0 = 4 * c + half * 2;          // always even -> 16B aligned
        const int4 q = *reinterpret_cast<const int4*>(&sbuf[idx0 * 2]);
        const int   jk0 = q.x;                      // j0 * K_SETS
        const float r0  = __int_as_float(q.y);
        const int   jk1 = q.z;                      // j1 * K_SETS
        const float r1  = __int_as_float(q.w);

        const float i0 = indT[jk0 + kcol];          // coalesced 64B per half
        const float i1 = indT[jk1 + kcol];

        const float x0 = fminf(fmaxf(r0 * i0, 1e-8f), 10000.0f);
        const float x1 = fminf(fmaxf(r1 * i1, 1e-8f), 10000.0f);
        const float w0 = fourth_root(x0);
        const float w1 = fourth_root(x1);
        const float n0 = (i0 < 0.1f) ? 1.f : 0.f;
        const float n1 = (i1 < 0.1f) ? 1.f : 0.f;

        const v2f aw = {w0, w1};
        const v2f an = {n0, n1};
        const v2f bf = { fmaf(coef0, sel1, sel0), fmaf(coef1, sel1, sel0) };

        dpos = __builtin_amdgcn_wmma_f32_16x16x4_f32(
            false, aw, false, bf, (short)0, dpos, false, false);
        dneg = __builtin_amdgcn_wmma_f32_16x16x4_f32(
            false, an, false, bf, (short)0, dneg, false, false);

        coef0 -= 4.f; coef1 -= 4.f;
    }

    // Fold this wave's partial sums into global accumulators.
    if (m < 2) {
        float* P1 = acc;
        float* P2 = acc + 1 * B_BATCH * K_SETS;
        float* N1 = acc + 2 * B_BATCH * K_SETS;
        float* N2 = acc + 3 * B_BATCH * K_SETS;
        #pragma unroll
        for (int v = 0; v < 8; ++v) {
            const int k = k0 + v + half * 8;
            const float pv = dpos[v], nv = dneg[v];
            if (m == 0) {
                atomicAdd(&P1[b * K_SETS + k], pv);
                atomicAdd(&N1[b * K_SETS + k], nv);
            } else {
                atomicAdd(&P2[b * K_SETS + k], pv);
                atomicAdd(&N2[b * K_SETS + k], nv);
            }
        }
    }
}

// ---------------------------------------------------------------------------
// Finalize es[b,k] from the four reductions.
// ---------------------------------------------------------------------------
__global__ void k_final(const float* __restrict__ acc, float* __restrict__ out) {
    const int i = blockIdx.x * blockDim.x + threadIdx.x;
    const int NK = B_BATCH * K_SETS;
    if (i >= NK) return;
    const float p1 = acc[i], p2 = acc[i + NK];
    const float n1 = acc[i + 2 * NK], n2 = acc[i + 3 * NK];
    const float sp = (p1 > 1e-8f) ? p2 / (p1 + 1e-10f) : 0.f;
    const float sn = (n1 > 1e-8f) ? n2 / (n1 + 1e-10f) : 0.f;
    out[i] = (sp - sn) * (1.0f / (float)G_GENES);
}

// ---------------------------------------------------------------------------
extern "C" void kernel_launch(void* const* d_in, const int* in_sizes, int n_in,
                              void* d_out, int out_size, void* d_ws, size_t ws_size,
                              hipStream_t stream) {
    const float* R  = (const float*)d_in[0];   // (16, 20000)
    const float* M  = (const float*)d_in[1];   // (128, 20000)
    const int*   S  = (const int*)d_in[2];     // (16, 20000)
    float*       out = (float*)d_out;          // (16, 128)

    // Workspace layout (floats): [avg: 128][indT: G*K][acc: 4*B*K]
    float* avg  = (float*)d_ws;
    float* indT = avg + 128;
    float* acc  = indT + (size_t)G_GENES * K_SETS;
    const int nacc = 4 * B_BATCH * K_SETS;

    k_avg<<<K_SETS, 256, 0, stream>>>(M, avg);
    k_build<<<dim3(G_GENES / 32, K_SETS / 32), dim3(32, 8), 0, stream>>>(M, avg, indT);
    k_zero<<<(nacc + 255) / 256, 256, 0, stream>>>(acc, nacc);
    k_main<<<dim3(G_GENES / GTILE, B_BATCH), 256, 0, stream>>>(R, S, indT, acc);
    k_final<<<(B_BATCH * K_SETS + 255) / 256, 256, 0, stream>>>(acc, out);
}